// RGCN_66262755442784
// MI455X (gfx1250) — compile-verified
//
#include <hip/hip_runtime.h>

#define R_REL 20
#define B_BASES 8
#define F 32

typedef __attribute__((ext_vector_type(2))) float v2f;
typedef __attribute__((ext_vector_type(8))) float v8f;

// Packed B-operand layout for V_WMMA_F32_16X16X4_F32:
//   Wp[(i>>1)*64 + o*2 + (i&1)] = W[i][o]
// so a lane's two K-values (VGPR x: K=k0, VGPR y: K=k0+1) are one aligned
// 8-byte load, coalesced across the 16 column lanes.
__device__ __forceinline__ int pack_idx(int i, int o) {
    return ((i >> 1) << 6) + (o << 1) + (i & 1);
}

// ---------------------------------------------------------------------------
// Kernel 1: W[r] = sum_b a[r,b] * V[b] for both layers, stored packed; also
// zero relation counters (runs first in the stream each call).
// ---------------------------------------------------------------------------
__global__ void build_w_kernel(const float* __restrict__ V1, const float* __restrict__ a1,
                               const float* __restrict__ V2, const float* __restrict__ a2,
                               float* __restrict__ W1, float* __restrict__ W2,
                               int* __restrict__ cnt) {
    if (blockIdx.x == 0 && threadIdx.x < R_REL) cnt[threadIdx.x] = 0;
    int l = blockIdx.x / R_REL;
    int r = blockIdx.x % R_REL;
    const float* V = l ? V2 : V1;
    const float* a = l ? a2 : a1;
    float*       W = l ? W2 : W1;
    for (int idx = threadIdx.x; idx < F * F; idx += blockDim.x) {
        int i = idx >> 5, o = idx & 31;
        float s = 0.f;
#pragma unroll
        for (int b = 0; b < B_BASES; ++b)
            s += a[r * B_BASES + b] * V[b * F * F + idx];
        W[r * F * F + pack_idx(i, o)] = s;
    }
}

// ---------------------------------------------------------------------------
// Kernel 1b: repack the self-loop weight matrices into WMMA B layout
// ---------------------------------------------------------------------------
__global__ void pack_loop_kernel(const float* __restrict__ loop1, const float* __restrict__ loop2,
                                 float* __restrict__ lp1, float* __restrict__ lp2) {
    const float* ssrc = blockIdx.x ? loop2 : loop1;
    float*       ddst = blockIdx.x ? lp2 : lp1;
    for (int idx = threadIdx.x; idx < F * F; idx += blockDim.x) {
        int i = idx >> 5, o = idx & 31;
        ddst[pack_idx(i, o)] = ssrc[idx];
    }
}

// ---------------------------------------------------------------------------
// Kernel 2: per-relation histogram (LDS-aggregated)
// ---------------------------------------------------------------------------
__global__ void count_rel_kernel(const int* __restrict__ ety, int E, int* __restrict__ cnt) {
    __shared__ int h[R_REL];
    if (threadIdx.x < R_REL) h[threadIdx.x] = 0;
    __syncthreads();
    for (int e = blockIdx.x * blockDim.x + threadIdx.x; e < E; e += gridDim.x * blockDim.x)
        atomicAdd(&h[ety[e]], 1);
    __syncthreads();
    if (threadIdx.x < R_REL) atomicAdd(&cnt[threadIdx.x], h[threadIdx.x]);
}

// ---------------------------------------------------------------------------
// Kernel 3: scan (R=20, single thread). cnt[] becomes the scatter cursor.
// ---------------------------------------------------------------------------
__global__ void scan_rel_kernel(int* __restrict__ cnt, int* __restrict__ start,
                                int* __restrict__ tileStart) {
    if (blockIdx.x == 0 && threadIdx.x == 0) {
        int run = 0, tt = 0;
        for (int r = 0; r < R_REL; ++r) {
            int c = cnt[r];
            start[r] = run;
            tileStart[r] = tt;
            cnt[r] = run;           // cursor for scatter
            run += c;
            tt += (c + 15) >> 4;
        }
        start[R_REL] = run;
        tileStart[R_REL] = tt;
    }
}

// ---------------------------------------------------------------------------
// Kernel 4: scatter edge ids into relation-sorted order[] (block-aggregated)
// ---------------------------------------------------------------------------
__global__ void scatter_rel_kernel(const int* __restrict__ ety, int E,
                                   int* __restrict__ cursor, int* __restrict__ order) {
    __shared__ int lcnt[R_REL];
    __shared__ int lbase[R_REL];
    if (threadIdx.x < R_REL) lcnt[threadIdx.x] = 0;
    __syncthreads();
    int e = blockIdx.x * blockDim.x + threadIdx.x;
    int r = 0, rank = 0;
    if (e < E) {
        r = ety[e];
        rank = atomicAdd(&lcnt[r], 1);
    }
    __syncthreads();
    if (threadIdx.x < R_REL) lbase[threadIdx.x] = atomicAdd(&cursor[threadIdx.x], lcnt[threadIdx.x]);
    __syncthreads();
    if (e < E) order[lbase[r] + rank] = e;
}

// ---------------------------------------------------------------------------
// Kernel 5: self-loop GEMM  hout = (relu?) xin @ wl + bias  via f32 WMMA.
// wl is in packed B layout. One wave per 16-node tile. RELU is compile-time.
// ---------------------------------------------------------------------------
template <int RELU>
__global__ void selfloop_wmma_kernel(const float* __restrict__ xin, const float* __restrict__ wl,
                                     const float* __restrict__ bias, float* __restrict__ hout,
                                     int ntiles) {
    int wave = blockIdx.x * (blockDim.x >> 5) + (threadIdx.x >> 5);
    if (wave >= ntiles) return;                    // wave-uniform: EXEC stays all-1s
    int lane = threadIdx.x & 31;
    int m  = lane & 15;                            // A row / B,C column
    int hi = lane >> 4;                            // lane half
    const v2f* xrow = (const v2f*)(xin + (size_t)(wave * 16 + m) * F) + hi;
    const v2f* wp   = (const v2f*)wl;

    v8f c0 = {0, 0, 0, 0, 0, 0, 0, 0};
    v8f c1 = {0, 0, 0, 0, 0, 0, 0, 0};
#pragma unroll
    for (int ks = 0; ks < 8; ++ks) {
        int p = ks * 2 + hi;                       // K-pair index = k0/2
        v2f a = xrow[ks * 2];                      // {x[k0], x[k0+1]}
        if (RELU) { a.x = fmaxf(a.x, 0.f); a.y = fmaxf(a.y, 0.f); }
        v2f b0 = wp[p * 32 + m];
        v2f b1 = wp[p * 32 + 16 + m];
        c0 = __builtin_amdgcn_wmma_f32_16x16x4_f32(false, a, false, b0, (short)0, c0, false, false);
        c1 = __builtin_amdgcn_wmma_f32_16x16x4_f32(false, a, false, b1, (short)0, c1, false, false);
    }

    int half = hi << 3;                            // C rows: M = v + 8*(lane/16)
    float bias0 = bias[m], bias1 = bias[16 + m];
#pragma unroll
    for (int v = 0; v < 8; ++v) {
        int row = wave * 16 + half + v;
        hout[(size_t)row * F + m]      = c0[v] + bias0;
        hout[(size_t)row * F + 16 + m] = c1[v] + bias1;
    }
}

// ---------------------------------------------------------------------------
// Kernel 6: edge messages via f32 WMMA. One wave = 16 edges of one relation.
// A = gathered x[src] rows (16x32), B = packed W[r] (32x32, two col tiles),
// D rows scatter-accumulated into h[dst] with hardware f32 atomics.
// Tail rows contribute exact 0.0 (A row zeroed); their atomics hit a valid
// slot -> harmless adds of zero.
// ---------------------------------------------------------------------------
template <int RELU>
__global__ void edge_wmma_kernel(const float* __restrict__ hin, const float* __restrict__ W,
                                 const int* __restrict__ src, const int* __restrict__ dst,
                                 const int* __restrict__ order, const int* __restrict__ start,
                                 const int* __restrict__ tileStart, float* __restrict__ hout) {
    int wave = blockIdx.x * (blockDim.x >> 5) + (threadIdx.x >> 5);
    int total = tileStart[R_REL];
    if (wave >= total) return;                     // wave-uniform

    // find relation of this tile (R=20, wave-uniform scalar scan)
    int r = 0;
#pragma unroll 1
    while (r < R_REL - 1 && wave >= tileStart[r + 1]) ++r;

    int segBase = start[r];
    int segEnd  = start[r + 1];
    int slot0   = segBase + (wave - tileStart[r]) * 16;

    int lane = threadIdx.x & 31;
    int m  = lane & 15;
    int hi = lane >> 4;

    // A-side edge for this lane's matrix row
    int  idxA   = slot0 + m;
    bool validA = idxA < segEnd;
    int  eA     = order[validA ? idxA : segBase];
    const v2f* xrow = (const v2f*)(hin + (size_t)src[eA] * F) + hi;

    // Preload B = packed W[r] column tiles into registers (L2-hot, 80KB total)
    const v2f* wp = (const v2f*)(W + r * (F * F));
    v2f bt0[8], bt1[8];
#pragma unroll
    for (int ks = 0; ks < 8; ++ks) {
        int p = ks * 2 + hi;
        bt0[ks] = wp[p * 32 + m];
        bt1[ks] = wp[p * 32 + 16 + m];
    }

    v8f c0 = {0, 0, 0, 0, 0, 0, 0, 0};
    v8f c1 = {0, 0, 0, 0, 0, 0, 0, 0};
#pragma unroll
    for (int ks = 0; ks < 8; ++ks) {
        v2f a = xrow[ks * 2];
        if (RELU) { a.x = fmaxf(a.x, 0.f); a.y = fmaxf(a.y, 0.f); }
        if (!validA) { a.x = 0.f; a.y = 0.f; }     // tail rows -> exact zero messages
        c0 = __builtin_amdgcn_wmma_f32_16x16x4_f32(false, a, false, bt0[ks], (short)0, c0, false, false);
        c1 = __builtin_amdgcn_wmma_f32_16x16x4_f32(false, a, false, bt1[ks], (short)0, c1, false, false);
    }

    // Scatter: lane covers D rows M = v + 8*(lane/16), column = m (and m+16)
    int half = hi << 3;
#pragma unroll
    for (int v = 0; v < 8; ++v) {
        int  slot = slot0 + half + v;
        bool ok   = slot < segEnd;
        int  e    = order[ok ? slot : segBase];
        float* hrow = hout + (size_t)dst[e] * F;   // invalid rows add 0.0f
        unsafeAtomicAdd(&hrow[m],      c0[v]);
        unsafeAtomicAdd(&hrow[16 + m], c1[v]);
    }
}

// ---------------------------------------------------------------------------
// Launch
// ---------------------------------------------------------------------------
extern "C" void kernel_launch(void* const* d_in, const int* in_sizes, int n_in,
                              void* d_out, int out_size, void* d_ws, size_t ws_size,
                              hipStream_t stream) {
    (void)n_in; (void)out_size; (void)ws_size;
    const float* x     = (const float*)d_in[0];
    const int*   src   = (const int*)d_in[1];
    const int*   dst   = (const int*)d_in[2];
    const int*   ety   = (const int*)d_in[3];
    const float* V1    = (const float*)d_in[4];
    const float* a1    = (const float*)d_in[5];
    const float* loop1 = (const float*)d_in[6];
    const float* b1    = (const float*)d_in[7];
    const float* V2    = (const float*)d_in[8];
    const float* a2    = (const float*)d_in[9];
    const float* loop2 = (const float*)d_in[10];
    const float* b2    = (const float*)d_in[11];
    float* out = (float*)d_out;

    const int Nn = in_sizes[0] / F;   // 100000
    const int Ee = in_sizes[1];       // 1600000

    // Workspace layout (all float sub-buffers 8-byte aligned)
    float* wsf = (float*)d_ws;
    float* W1  = wsf;                          // R*F*F
    float* W2  = W1 + R_REL * F * F;           // R*F*F
    float* lp1 = W2 + R_REL * F * F;           // F*F packed loop1
    float* lp2 = lp1 + F * F;                  // F*F packed loop2
    float* h1  = lp2 + F * F;                  // N*F
    int* wsi       = (int*)(h1 + (size_t)Nn * F);
    int* cnt       = wsi;                      // R       (histogram -> cursor)
    int* start     = cnt + R_REL;              // R+1     (edge segment offsets)
    int* tileStart = start + R_REL + 1;        // R+1     (wmma tile offsets)
    int* order     = tileStart + R_REL + 1;    // E       (relation-sorted edge ids)

    const int ntiles   = Nn / 16;                        // 6250, exact
    const int slBlocks = (ntiles + 7) / 8;               // 8 waves / block
    const int maxTiles = Ee / 16 + R_REL + 1;
    const int edBlocks = (maxTiles + 7) / 8;
    const int scBlocks = (Ee + 255) / 256;

    // Weights (packed) + edge sort by relation (shared by both layers)
    build_w_kernel<<<2 * R_REL, 256, 0, stream>>>(V1, a1, V2, a2, W1, W2, cnt);
    pack_loop_kernel<<<2, 256, 0, stream>>>(loop1, loop2, lp1, lp2);
    count_rel_kernel<<<1024, 256, 0, stream>>>(ety, Ee, cnt);
    scan_rel_kernel<<<1, 32, 0, stream>>>(cnt, start, tileStart);
    scatter_rel_kernel<<<scBlocks, 256, 0, stream>>>(ety, Ee, cnt, order);

    // Layer 1: h1 = x @ loop1 + b1  +  scatter(x[src] @ W1[r] -> dst)
    selfloop_wmma_kernel<0><<<slBlocks, 256, 0, stream>>>(x, lp1, b1, h1, ntiles);
    edge_wmma_kernel<0><<<edBlocks, 256, 0, stream>>>(x, W1, src, dst, order, start, tileStart, h1);

    // Layer 2 (ReLU fused into loads of h1): out = relu(h1) @ loop2 + b2 + messages
    selfloop_wmma_kernel<1><<<slBlocks, 256, 0, stream>>>(h1, lp2, b2, out, ntiles);
    edge_wmma_kernel<1><<<edBlocks, 256, 0, stream>>>(h1, W2, src, dst, order, start, tileStart, out);
}